// Wkv7_75316546503072
// MI455X (gfx1250) — compile-verified
//
#include <hip/hip_runtime.h>
#include <stdint.h>

#define NUM_HEADS 32
#define HEAD 64
#define SEQ 4096
#define D_MODEL (NUM_HEADS*HEAD)   // 2048
#define CHUNK 16
#define NCHUNK (SEQ/CHUNK)         // 256
#define THREADS 512

// padded LDS row strides (even -> 8B-aligned v2f pairs; non-multiple-of-64 -> no bank walls)
#define SP 66   // Slds row stride
#define GP 66   // Gt row stride (32 rows x 64)
#define HP 34   // Htt row stride (64 rows x 32)
#define VP 18   // Vt/SAv/X0 row stride (64 rows x 16)
#define DP 18   // Dmt row stride (16 rows x 16)

typedef float v2f __attribute__((ext_vector_type(2)));
typedef float v8f __attribute__((ext_vector_type(8)));
typedef unsigned int u32x4 __attribute__((ext_vector_type(4)));
typedef int i32x4 __attribute__((ext_vector_type(4)));
typedef int i32x8 __attribute__((ext_vector_type(8)));

#if defined(__AMDGCN__) && __has_builtin(__builtin_amdgcn_tensor_load_to_lds) && \
    __has_builtin(__builtin_amdgcn_s_wait_tensorcnt)
#define HAVE_TDM 1
#else
#define HAVE_TDM 0
#endif

#if HAVE_TDM
__device__ __forceinline__ unsigned lds_byte_off(const void* p) {
  return (unsigned)(uintptr_t)(__attribute__((address_space(3))) const void*)p;
}
// TDM D# for a 16x64 f32 tile out of a (rows x 2048) f32 tensor (CDNA5 ISA)
__device__ __forceinline__ void tdm_load_tile(const float* gsrc, unsigned lds_byte) {
  unsigned long long ga = (unsigned long long)(uintptr_t)gsrc;
  u32x4 g0;
  g0[0] = 1u;                                           // count=1, user mode
  g0[1] = lds_byte;                                     // lds_addr (bytes)
  g0[2] = (unsigned)ga;                                 // global_addr[31:0]
  g0[3] = (unsigned)((ga >> 32) & 0x1FFFFFFu) | (2u << 30); // addr[56:32] | type=2
  i32x8 g1;
  g1[0] = 0x00020000;            // workgroup_mask=0, data_size=2 (4B), no barrier/iter/pad
  g1[1] = (int)(2048u << 16);    // atomic_barrier_addr=0 | tensor_dim0 lo16 = 2048
  g1[2] = (int)(4096u << 16);    // tensor_dim0 hi=0     | tensor_dim1 lo16 = 4096
  g1[3] = (int)(64u << 16);      // tensor_dim1 hi=0     | tile_dim0 = 64
  g1[4] = 16;                    // tile_dim1 = 16, tile_dim2 = 0
  g1[5] = 2048;                  // tensor_dim0_stride lo32 = 2048
  g1[6] = 0; g1[7] = 0;          // stride hi, tensor_dim1_stride = 0
  i32x4 z4 = {0, 0, 0, 0};
#if __has_include(<hip/amd_detail/amd_gfx1250_TDM.h>)
  i32x8 z8 = {0, 0, 0, 0, 0, 0, 0, 0};   // clang-23 therock lane: 6-arg builtin
  __builtin_amdgcn_tensor_load_to_lds(g0, g1, z4, z4, z8, 0);
#else
  __builtin_amdgcn_tensor_load_to_lds(g0, g1, z4, z4, 0);  // ROCm 7.2: 5-arg builtin
#endif
}
#endif

__device__ __forceinline__ float allreduce_add(float v) {
#pragma unroll
  for (int m = 16; m >= 1; m >>= 1) v += __shfl_xor(v, m, 32);
  return v;
}

// One block per head. Chunked linear-recurrence wkv7:
//   S_t = S_{t-1}(diag(w_t)+a_t b_t^T) + v_t k_t^T ,  x_t = S_t r_t
// Per 16-step chunk: factored chains -> rtil/atil/khat/btil/D, then f32 WMMA for
// S@[Atil|Rtil], X0+V@D (outputs), and the exact rank-32 state update.
__global__ __launch_bounds__(THREADS, 1)
void wkv7_chunked(const float* __restrict__ R, const float* __restrict__ W,
                  const float* __restrict__ K, const float* __restrict__ V,
                  const float* __restrict__ A, const float* __restrict__ B,
                  const float* __restrict__ S0, float* __restrict__ out) {
  __shared__ float Slds[HEAD*SP];          // state, padded rows
  __shared__ float IN[2][6*CHUNK*HEAD];    // double-buffered chunk inputs (TDM dest)
  __shared__ float Gt[2*CHUNK*GP];         // [n][j]: n<16 atil_n, n>=16 rtil_{n-16}
  __shared__ float Htt[HEAD*HP];           // [j][s]: s<16 b_s*Wsuf_s, s>=16 khat_{s-16}
  __shared__ float SAv[HEAD*VP];           // S @ Atil (64x16)
  __shared__ float X0[HEAD*VP];            // S @ Rtil (64x16)
  __shared__ float Vt[HEAD*VP];            // V transposed: Vt[m][s] = v_s[m]
  __shared__ float Dmt[CHUNK*DP];          // Dmt[i][s] = k_s . P(s+1..i) r_i
  __shared__ float Wc[HEAD];               // full-chunk column decay product

  const int tid = threadIdx.x, wid = tid >> 5, lane = tid & 31;
  const int half = lane >> 4, lr = lane & 15;
  const int head = blockIdx.x;
  float* outX = out;                       // (T,H,1,N) f32
  float* outS = out + SEQ*D_MODEL;         // (H,N,N)  f32
  const float* srcs[6] = {R, W, K, V, A, B};

  for (int idx = tid; idx < HEAD*HEAD; idx += THREADS)
    Slds[(idx >> 6)*SP + (idx & 63)] = S0[head*HEAD*HEAD + idx];

#if HAVE_TDM
  if (wid == 0) {                          // prologue: DMA chunk 0 into buffer 0
#pragma unroll
    for (int q = 0; q < 6; ++q)
      tdm_load_tile(srcs[q] + head*HEAD, lds_byte_off(&IN[0][0]) + q*CHUNK*HEAD*4);
  }
#endif

  for (int ch = 0; ch < NCHUNK; ++ch) {
    const int t0 = ch*CHUNK;
    float* cur = &IN[ch & 1][0];
    const float *rB = cur, *wB = cur + 1024, *kB = cur + 2048,
                *vB = cur + 3072, *aB = cur + 4096, *bB = cur + 5120;

#if HAVE_TDM
    if (wid == 0) __builtin_amdgcn_s_wait_tensorcnt((short)0);  // chunk ch landed
    __syncthreads();                        // data visible; prev chunk fully consumed
    if (wid == 0 && ch + 1 < NCHUNK) {      // DMA next chunk while we compute
      float* nxt = &IN[(ch + 1) & 1][0];
#pragma unroll
      for (int q = 0; q < 6; ++q)
        tdm_load_tile(srcs[q] + (t0 + CHUNK)*D_MODEL + head*HEAD,
                      lds_byte_off(nxt) + q*CHUNK*HEAD*4);
    }
#else
    __syncthreads();                        // prev chunk fully consumed
    for (int q = 0; q < 6; ++q) {
      for (int idx = tid; idx < CHUNK*HEAD; idx += THREADS) {
        const int g = (t0 + (idx >> 6))*D_MODEL + head*HEAD + (idx & 63);
        cur[q*1024 + idx] = srcs[q][g];
        if (ch + 1 < NCHUNK && (idx & 31) == 0)
          __builtin_prefetch(&srcs[q][g + CHUNK*D_MODEL], 0, 0);
      }
    }
    __syncthreads();
#endif
    // prep: zero D, build transposed V copy (vectorizable WMMA A-operand)
    for (int idx = tid; idx < CHUNK*DP; idx += THREADS) Dmt[idx] = 0.f;
    for (int idx = tid; idx < CHUNK*HEAD; idx += THREADS)
      Vt[(idx & 63)*VP + (idx >> 6)] = vB[idx];
    __syncthreads();

    // ---- phase A: factored transition chains; wave 'wid' owns chain i=wid ----
    {
      const int i = wid;
      // r-chain: rtil_i = A_0..A_i r_i ; record d[i][m] before each apply
      float ulo = rB[i*64 + lane], uhi = rB[i*64 + 32 + lane];
      for (int m = i; m >= 0; --m) {
        float kd = kB[m*64+lane]*ulo + kB[m*64+32+lane]*uhi;
        float bd = bB[m*64+lane]*ulo + bB[m*64+32+lane]*uhi;
        kd = allreduce_add(kd); bd = allreduce_add(bd);
        if (lane == 0) Dmt[i*DP + m] = kd;
        ulo = wB[m*64+lane]*ulo + aB[m*64+lane]*bd;
        uhi = wB[m*64+32+lane]*uhi + aB[m*64+32+lane]*bd;
      }
      Gt[(16 + i)*GP + lane] = ulo;
      Gt[(16 + i)*GP + 32 + lane] = uhi;
      // a-chain: atil_i = A_0..A_{i-1} a_i
      ulo = aB[i*64+lane]; uhi = aB[i*64+32+lane];
      for (int m = i - 1; m >= 0; --m) {
        float bd = allreduce_add(bB[m*64+lane]*ulo + bB[m*64+32+lane]*uhi);
        ulo = wB[m*64+lane]*ulo + aB[m*64+lane]*bd;
        uhi = wB[m*64+32+lane]*uhi + aB[m*64+32+lane]*bd;
      }
      Gt[i*GP + lane] = ulo; Gt[i*GP + 32 + lane] = uhi;
      // k-chain: khat_i^T = k_i^T A_{i+1}..A_{15}
      ulo = kB[i*64+lane]; uhi = kB[i*64+32+lane];
      for (int m = i + 1; m < CHUNK; ++m) {
        float ad = allreduce_add(aB[m*64+lane]*ulo + aB[m*64+32+lane]*uhi);
        ulo = wB[m*64+lane]*ulo + bB[m*64+lane]*ad;
        uhi = wB[m*64+32+lane]*uhi + bB[m*64+32+lane]*ad;
      }
      Htt[lane*HP + 16 + i] = ulo; Htt[(lane + 32)*HP + 16 + i] = uhi;
      // decay suffix scan: btil_s = b_s * prod_{m>s} w_m ; Wc = prod_all w
      if (wid == 0) {
        float plo = 1.f, phi = 1.f;
        for (int s = CHUNK - 1; s >= 0; --s) {
          Htt[lane*HP + s]        = bB[s*64+lane]    * plo;
          Htt[(lane + 32)*HP + s] = bB[s*64+32+lane] * phi;
          plo *= wB[s*64+lane]; phi *= wB[s*64+32+lane];
        }
        Wc[lane] = plo; Wc[32 + lane] = phi;
      }
    }
    __syncthreads();

    // ---- T1: [SA | X0] = S @ [Atil|Rtil]  (64x64 @ 64x32, f32 WMMA) ----
    if (wid < 8) {
      const int tr = wid >> 1, tc = wid & 1;
      v8f c = {};
#pragma unroll
      for (int kk = 0; kk < 64; kk += 4) {
        v2f av = *(const v2f*)&Slds[(tr*16 + lr)*SP + kk + 2*half];
        v2f bv = *(const v2f*)&Gt[(tc*16 + lr)*GP + kk + 2*half];
        c = __builtin_amdgcn_wmma_f32_16x16x4_f32(false, av, false, bv,
                                                  (short)0, c, false, false);
      }
      float* dst = (tc == 0) ? SAv : X0;
#pragma unroll
      for (int vv = 0; vv < 8; ++vv)
        dst[(tr*16 + vv + 8*half)*VP + lr] = c[vv];
    }
    __syncthreads();

    // ---- T2 (waves 0-3): X = X0 + V @ D -> write x to global ----
    if (wid < 4) {
      const int tr = wid;
      v8f c;
#pragma unroll
      for (int vv = 0; vv < 8; ++vv) c[vv] = X0[(tr*16 + vv + 8*half)*VP + lr];
#pragma unroll
      for (int kk = 0; kk < 16; kk += 4) {
        v2f av = *(const v2f*)&Vt[(tr*16 + lr)*VP + kk + 2*half];
        v2f bv = *(const v2f*)&Dmt[lr*DP + kk + 2*half];
        c = __builtin_amdgcn_wmma_f32_16x16x4_f32(false, av, false, bv,
                                                  (short)0, c, false, false);
      }
#pragma unroll
      for (int vv = 0; vv < 8; ++vv)
        outX[(t0 + lr)*D_MODEL + head*HEAD + tr*16 + vv + 8*half] = c[vv];
    }

    // ---- T3 (all 16 waves): S <- S*diag(Wc) + [SA|V] @ [Btil|Khat]^T ----
    {
      const int tr = wid >> 2, tc = wid & 3;
      v8f c;
#pragma unroll
      for (int vv = 0; vv < 8; ++vv)
        c[vv] = Slds[(tr*16 + vv + 8*half)*SP + tc*16 + lr] * Wc[tc*16 + lr];
#pragma unroll
      for (int kk = 0; kk < 32; kk += 4) {
        const int s0 = kk + 2*half;
        v2f av, bv;
        if (kk < 16) av = *(const v2f*)&SAv[(tr*16 + lr)*VP + s0];
        else         av = *(const v2f*)&Vt[(tr*16 + lr)*VP + s0 - 16];
        bv = *(const v2f*)&Htt[(tc*16 + lr)*HP + s0];
        c = __builtin_amdgcn_wmma_f32_16x16x4_f32(false, av, false, bv,
                                                  (short)0, c, false, false);
      }
      __syncthreads();   // everyone done reading Slds/SAv/Vt/Htt before overwrite
#pragma unroll
      for (int vv = 0; vv < 8; ++vv)
        Slds[(tr*16 + vv + 8*half)*SP + tc*16 + lr] = c[vv];
    }
  }

  __syncthreads();
  for (int idx = tid; idx < HEAD*HEAD; idx += THREADS)
    outS[head*HEAD*HEAD + idx] = Slds[(idx >> 6)*SP + (idx & 63)];
}

extern "C" void kernel_launch(void* const* d_in, const int* in_sizes, int n_in,
                              void* d_out, int out_size, void* d_ws, size_t ws_size,
                              hipStream_t stream) {
  // inputs: [0]=seq_length(int), [1]=r, [2]=w, [3]=k, [4]=v, [5]=a, [6]=b, [7]=state2
  const float* R  = (const float*)d_in[1];
  const float* W  = (const float*)d_in[2];
  const float* K  = (const float*)d_in[3];
  const float* V  = (const float*)d_in[4];
  const float* A  = (const float*)d_in[5];
  const float* B  = (const float*)d_in[6];
  const float* S0 = (const float*)d_in[7];
  float* out = (float*)d_out;
  (void)in_sizes; (void)n_in; (void)out_size; (void)d_ws; (void)ws_size;
  wkv7_chunked<<<NUM_HEADS, THREADS, 0, stream>>>(R, W, K, V, A, B, S0, out);
}